// LlamaAttention_82695300317687
// MI455X (gfx1250) — compile-verified
//
#include <hip/hip_runtime.h>
#include <hip/hip_bf16.h>
#include <math.h>

#define HID   4096
#define NH    32
#define NKVH  8
#define DHEAD 128
#define SEQ   64
#define BS    16
#define MC    1024
#define GRP   (NH / NKVH)                  // 4 query heads per kv head
#define QKVC  ((NH + 2 * NKVH) * DHEAD)    // 6144
#define KSPLIT 8

typedef float v2f __attribute__((ext_vector_type(2)));
typedef float v8f __attribute__((ext_vector_type(8)));

// ---------------------------------------------------------------------------
// Split-K GEMM partial: Cpart[z][64,N] = A[64,Kz] * W[N,Kz]^T (fp32 WMMA).
// One wave owns all 64 output rows x (16*NT) columns for one K-slice of
// K/KSPLIT, so each weight element is loaded exactly once across the grid
// while grid parallelism is KSPLIT x higher than a monolithic-K version
// (needed to reach the 23.3 TB/s weight-stream roofline; a single wave is
// latency-bound at ~10 GB/s).  K is processed in chunks of 8 with two
// explicit register buffers so the next chunk's loads are in flight during
// the 16 WMMAs of the current chunk.
//
// Fragment layouts (ISA 7.12.2):
//   A 16x4 : lane(lo,hi), vgpr v -> A[M=lo][K = 2*hi + v]
//   B 4x16 : lane(lo,hi), vgpr v -> B[K = 2*hi + v][N=lo] = W[N=lo][K=...]
//   D      : vgpr j -> C[M = j + 8*hi][N = lo]
// ---------------------------------------------------------------------------
template <int NT>
__global__ __launch_bounds__(32) void wmma_gemm_splitk(
    const float* __restrict__ A, const float* __restrict__ W,
    float* __restrict__ Cpart, int N, int K)
{
    const int lane = threadIdx.x & 31;
    const int lo = lane & 15;
    const int hi = lane >> 4;
    const int n0 = blockIdx.x * (16 * NT);
    const int ks = K / KSPLIT;
    const int kbase = blockIdx.y * ks;
    const int kend = kbase + ks;

    v8f acc[4][NT];
#pragma unroll
    for (int r = 0; r < 4; ++r)
#pragma unroll
        for (int t = 0; t < NT; ++t) acc[r][t] = (v8f)0.0f;

    // loop-invariant 32-bit per-lane element offsets; the k-dependent part
    // stays uniform (SGPR) to favor SADDR-form global loads
    unsigned aoff[4];
#pragma unroll
    for (int r = 0; r < 4; ++r)
        aoff[r] = (unsigned)((r * 16 + lo) * K + 2 * hi);
    unsigned woff[NT];
#pragma unroll
    for (int t = 0; t < NT; ++t)
        woff[t] = (unsigned)((n0 + 16 * t + lo) * K + 2 * hi);

    // warm the streams (global_prefetch_b8)
#pragma unroll
    for (int r = 0; r < 4; ++r) __builtin_prefetch(A + kbase + aoff[r], 0, 3);
#pragma unroll
    for (int t = 0; t < NT; ++t) __builtin_prefetch(W + kbase + woff[t], 0, 3);

#define LOAD_CHUNK(kk, Ab, Bb)                                         \
    do {                                                               \
        const float* Ak = A + (kk);                                    \
        const float* Wk = W + (kk);                                    \
        _Pragma("unroll") for (int r = 0; r < 4; ++r) {                \
            Ab[r][0] = *(const v2f*)(Ak + aoff[r]);                    \
            Ab[r][1] = *(const v2f*)(Ak + aoff[r] + 4);                \
        }                                                              \
        _Pragma("unroll") for (int t = 0; t < NT; ++t) {               \
            Bb[t][0] = *(const v2f*)(Wk + woff[t]);                    \
            Bb[t][1] = *(const v2f*)(Wk + woff[t] + 4);                \
        }                                                              \
    } while (0)

#define COMPUTE_CHUNK(Ab, Bb)                                          \
    do {                                                               \
        _Pragma("unroll") for (int s = 0; s < 2; ++s)                  \
        _Pragma("unroll") for (int r = 0; r < 4; ++r)                  \
        _Pragma("unroll") for (int t = 0; t < NT; ++t)                 \
            acc[r][t] = __builtin_amdgcn_wmma_f32_16x16x4_f32(         \
                false, Ab[r][s], false, Bb[t][s], (short)0,            \
                acc[r][t], false, false);                              \
    } while (0)

    v2f a0[4][2], b0[NT][2], a1[4][2], b1[NT][2];
    LOAD_CHUNK(kbase, a0, b0);
    for (int k0 = kbase; k0 < kend; k0 += 16) {
        LOAD_CHUNK(k0 + 8, a1, b1);   // in-flight while computing chunk k0
        COMPUTE_CHUNK(a0, b0);
        if (k0 + 16 < kend) LOAD_CHUNK(k0 + 16, a0, b0);
        COMPUTE_CHUNK(a1, b1);
    }
#undef LOAD_CHUNK
#undef COMPUTE_CHUNK

    float* Cz = Cpart + (size_t)blockIdx.y * SEQ * N;
#pragma unroll
    for (int r = 0; r < 4; ++r)
#pragma unroll
        for (int t = 0; t < NT; ++t)
#pragma unroll
            for (int j = 0; j < 8; ++j) {
                int row = r * 16 + j + 8 * hi;
                Cz[(size_t)row * N + n0 + 16 * t + lo] = acc[r][t][j];
            }
}

// ---------------------------------------------------------------------------
// Deterministic fixed-order sum of the KSPLIT partial slabs.
// ---------------------------------------------------------------------------
__global__ __launch_bounds__(256) void reduce_splitk(
    const float* __restrict__ part, float* __restrict__ outp, int n)
{
    int i = blockIdx.x * 256 + threadIdx.x;
    if (i < n) {
        float s = 0.0f;
#pragma unroll
        for (int p = 0; p < KSPLIT; ++p) s += part[(size_t)p * n + i];
        outp[i] = s;
    }
}

// ---------------------------------------------------------------------------
// RoPE on q and k, split qkv into Q [S,NH,D], Kn [S,NKV,D], Vn [S,NKV,D].
// ---------------------------------------------------------------------------
__global__ __launch_bounds__(256) void rope_split_kernel(
    const float* __restrict__ qkv, const int* __restrict__ positions,
    float* __restrict__ Q, float* __restrict__ Kn, float* __restrict__ Vn)
{
    const int s = blockIdx.x;
    const float pos = (float)positions[s];
    const float* row = qkv + (size_t)s * QKVC;
    const float lntheta = __logf(10000.0f);

    for (int i = threadIdx.x; i < NH * (DHEAD / 2); i += blockDim.x) {
        int h = i >> 6, j = i & 63;
        float inv = __expf(-(2.0f * (float)j / (float)DHEAD) * lntheta);
        float ang = pos * inv;
        float c = cosf(ang), sn = sinf(ang);
        float x1 = row[h * DHEAD + j];
        float x2 = row[h * DHEAD + 64 + j];
        float* qo = Q + ((size_t)s * NH + h) * DHEAD;
        qo[j]      = x1 * c - x2 * sn;
        qo[64 + j] = x2 * c + x1 * sn;
    }
    for (int i = threadIdx.x; i < NKVH * (DHEAD / 2); i += blockDim.x) {
        int h = i >> 6, j = i & 63;
        float inv = __expf(-(2.0f * (float)j / (float)DHEAD) * lntheta);
        float ang = pos * inv;
        float c = cosf(ang), sn = sinf(ang);
        const float* kr = row + NH * DHEAD + h * DHEAD;
        float x1 = kr[j];
        float x2 = kr[64 + j];
        float* ko = Kn + ((size_t)s * NKVH + h) * DHEAD;
        ko[j]      = x1 * c - x2 * sn;
        ko[64 + j] = x2 * c + x1 * sn;
    }
    for (int i = threadIdx.x; i < NKVH * DHEAD; i += blockDim.x)
        Vn[(size_t)s * NKVH * DHEAD + i] = row[(NH + NKVH) * DHEAD + i];
}

// ---------------------------------------------------------------------------
// Paged grouped-query attention, one workgroup per (kv head, sequence).
// 8 waves stream context tokens with online softmax; lane owns a 4-float
// d-slice; scores reduced with full-wave32 shuffle butterflies.
// ---------------------------------------------------------------------------
__global__ __launch_bounds__(256) void attn_kernel(
    const float* __restrict__ Q, const float* __restrict__ Kn,
    const float* __restrict__ Vn, const float* __restrict__ k_cache,
    const float* __restrict__ v_cache, const int* __restrict__ block_tables,
    const int* __restrict__ context_lens, const int* __restrict__ positions,
    float* __restrict__ out)
{
    const int kv = blockIdx.x;
    const int s = blockIdx.y;
    const int ctx = context_lens[s];
    const int pos = positions[s];
    const int wave = threadIdx.x >> 5;
    const int lane = threadIdx.x & 31;

    __shared__ float qs[GRP][DHEAD];
    __shared__ float sm[8][GRP];
    __shared__ float sl[8][GRP];
    __shared__ float sacc[8][GRP][DHEAD];

    const float scale = 0.08838834764831845f;  // 128^-0.5
    for (int i = threadIdx.x; i < GRP * DHEAD; i += blockDim.x) {
        int g = i >> 7, d = i & 127;
        qs[g][d] = Q[((size_t)s * NH + kv * GRP + g) * DHEAD + d] * scale;
    }
    __syncthreads();

    float4 qreg[GRP];
#pragma unroll
    for (int g = 0; g < GRP; ++g)
        qreg[g] = *(const float4*)(&qs[g][lane << 2]);

    float m[GRP], l[GRP];
    float4 acc[GRP];
#pragma unroll
    for (int g = 0; g < GRP; ++g) {
        m[g] = -INFINITY;
        l[g] = 0.0f;
        acc[g] = make_float4(0.f, 0.f, 0.f, 0.f);
    }

    const int* bt = block_tables + (size_t)s * (MC / BS);

    for (int t = wave; t < ctx; t += 8) {
        const float* kptr;
        const float* vptr;
        if (t == pos) {  // token being decoded: use freshly computed k/v
            kptr = Kn + ((size_t)s * NKVH + kv) * DHEAD;
            vptr = Vn + ((size_t)s * NKVH + kv) * DHEAD;
        } else {
            int blk = bt[t >> 4];
            size_t base = (((size_t)blk * BS + (t & 15)) * NKVH + kv) * DHEAD;
            kptr = k_cache + base;
            vptr = v_cache + base;
        }
        float4 kf = *(const float4*)(kptr + (lane << 2));
        float sc[GRP];
#pragma unroll
        for (int g = 0; g < GRP; ++g) {
            float p = qreg[g].x * kf.x + qreg[g].y * kf.y +
                      qreg[g].z * kf.z + qreg[g].w * kf.w;
#pragma unroll
            for (int o = 16; o > 0; o >>= 1) p += __shfl_xor(p, o, 32);
            sc[g] = p;
        }
        float4 vf = *(const float4*)(vptr + (lane << 2));
#pragma unroll
        for (int g = 0; g < GRP; ++g) {
            float nm = fmaxf(m[g], sc[g]);
            float resc = __expf(m[g] - nm);
            float p = __expf(sc[g] - nm);
            acc[g].x = acc[g].x * resc + p * vf.x;
            acc[g].y = acc[g].y * resc + p * vf.y;
            acc[g].z = acc[g].z * resc + p * vf.z;
            acc[g].w = acc[g].w * resc + p * vf.w;
            l[g] = l[g] * resc + p;
            m[g] = nm;
        }
    }

#pragma unroll
    for (int g = 0; g < GRP; ++g) {
        sm[wave][g] = m[g];
        sl[wave][g] = l[g];
        *(float4*)(&sacc[wave][g][lane << 2]) = acc[g];
    }
    __syncthreads();

    if (threadIdx.x < GRP * 32) {
        int g = threadIdx.x >> 5, ln = threadIdx.x & 31;
        float M = -INFINITY;
#pragma unroll
        for (int w = 0; w < 8; ++w) M = fmaxf(M, sm[w][g]);
        float L = 0.f, o0 = 0.f, o1 = 0.f, o2 = 0.f, o3 = 0.f;
#pragma unroll
        for (int w = 0; w < 8; ++w) {
            float f = __expf(sm[w][g] - M);
            L += sl[w][g] * f;
            o0 += sacc[w][g][(ln << 2) + 0] * f;
            o1 += sacc[w][g][(ln << 2) + 1] * f;
            o2 += sacc[w][g][(ln << 2) + 2] * f;
            o3 += sacc[w][g][(ln << 2) + 3] * f;
        }
        float invL = 1.0f / L;
        float* op = out + ((size_t)s * NH + kv * GRP + g) * DHEAD + (ln << 2);
        op[0] = o0 * invL;
        op[1] = o1 * invL;
        op[2] = o2 * invL;
        op[3] = o3 * invL;
    }
}

// ---------------------------------------------------------------------------
extern "C" void kernel_launch(void* const* d_in, const int* in_sizes, int n_in,
                              void* d_out, int out_size, void* d_ws,
                              size_t ws_size, hipStream_t stream)
{
    const float* hidden       = (const float*)d_in[0];
    const float* k_cache      = (const float*)d_in[1];
    const float* v_cache      = (const float*)d_in[2];
    const float* qkv_w        = (const float*)d_in[3];
    const float* o_w          = (const float*)d_in[4];
    const int*   positions    = (const int*)d_in[5];
    const int*   context_lens = (const int*)d_in[6];
    const int*   block_tables = (const int*)d_in[7];
    float* out = (float*)d_out;

    float* ws   = (float*)d_ws;
    float* qkv  = ws;                           // SEQ * QKVC
    float* Q    = qkv + (size_t)SEQ * QKVC;     // SEQ * NH * DHEAD
    float* Kn   = Q + (size_t)SEQ * NH * DHEAD; // SEQ * NKVH * DHEAD
    float* Vn   = Kn + (size_t)SEQ * NKVH * DHEAD;
    float* attn = Vn + (size_t)SEQ * NKVH * DHEAD;
    float* part = attn + (size_t)SEQ * NH * DHEAD;  // KSPLIT*SEQ*QKVC slab

    // 1) QKV projection (split-K partials + deterministic reduction)
    wmma_gemm_splitk<2><<<dim3(QKVC / 32, KSPLIT), 32, 0, stream>>>(
        hidden, qkv_w, part, QKVC, HID);
    reduce_splitk<<<(SEQ * QKVC + 255) / 256, 256, 0, stream>>>(
        part, qkv, SEQ * QKVC);

    // 2) RoPE + split into Q / Kn / Vn
    rope_split_kernel<<<SEQ, 256, 0, stream>>>(qkv, positions, Q, Kn, Vn);

    // 3) Paged GQA with online softmax
    attn_kernel<<<dim3(NKVH, SEQ), 256, 0, stream>>>(
        Q, Kn, Vn, k_cache, v_cache, block_tables, context_lens, positions,
        attn);

    // 4) Output projection (split-K partials + deterministic reduction)
    wmma_gemm_splitk<2><<<dim3(HID / 32, KSPLIT), 32, 0, stream>>>(
        attn, o_w, part, HID, HID);
    reduce_splitk<<<(SEQ * HID + 255) / 256, 256, 0, stream>>>(
        part, out, SEQ * HID);
}